// QModel_65481071403459
// MI455X (gfx1250) — compile-verified
//
#include <hip/hip_runtime.h>
#include <math.h>

// MI455X / gfx1250, wave32. One workgroup (256 thr = 8 waves) per batch item.
// 12-qubit state (4096 complex f32 = 32KB) resident in LDS (320KB/WGP).
// Fused 4-qubit RY blocks applied with v_wmma_f32_16x16x4_f32.

typedef float v2f __attribute__((ext_vector_type(2)));
typedef float v8f __attribute__((ext_vector_type(8)));

#define NW   12
#define QDIM 4096
#define NL   2

// Full 12-bit index from a group's 4-bit nibble and the remaining 8 bits `o`.
// Wire w <-> index bit (11-w). Group g covers wires 4g..4g+3.
__device__ __forceinline__ int qidx(int g, int nib, int o) {
  if (g == 0)      return (nib << 8) | o;                          // bits 11..8
  else if (g == 1) return ((o & 0xF0) << 4) | (nib << 4) | (o & 15); // bits 7..4
  else             return (o << 4) | nib;                          // bits 3..0
}

__global__ __launch_bounds__(256) void qsim_kernel(
    const float* __restrict__ sbatch,   // (1024, 4096) f32; only cols 0..11 used
    const float* __restrict__ params,   // (2, 12, 2) f32
    const float* __restrict__ head_w,   // (1, 12) f32
    const float* __restrict__ head_b,   // (1,) f32
    float* __restrict__ out)            // (1024,) f32
{
  __shared__ float st[2][QDIM];          // [0]=re, [1]=im
  __shared__ float Mmat[256];            // fused 16x16 RY block (current group)
  __shared__ float cry[NL*NW], sry[NL*NW], hz[NL*NW];
  __shared__ float ca[NW], sa[NW], hw[NW];
  __shared__ float red[256];

  const int tid = threadIdx.x;
  const int bid = blockIdx.x;

  // Per-block small trig tables (params are shared across the batch).
  if (tid < NL*NW) {
    float th = params[tid*2 + 0] * 0.5f;   // RY half-angle, index l*12+w
    cry[tid] = cosf(th); sry[tid] = sinf(th);
    hz[tid]  = params[tid*2 + 1] * 0.5f;   // RZ half-angle
  }
  if (tid < NW) {
    float a = sbatch[(size_t)bid * QDIM + tid] * 0.5f;
    ca[tid] = cosf(a); sa[tid] = sinf(a);
    hw[tid] = head_w[tid];
  }
  __syncthreads();

  // Initial 12 RYs on |0..0> give a product state: amp_i = prod_w (bit? sin : cos).
  float phb = 1.0f;
  #pragma unroll
  for (int w = 0; w < 8; ++w) phb *= ((tid >> (7 - w)) & 1) ? sa[w] : ca[w];
  #pragma unroll
  for (int j = 0; j < 16; ++j) {
    float p = phb;
    #pragma unroll
    for (int w = 8; w < NW; ++w) p *= ((j >> (11 - w)) & 1) ? sa[w] : ca[w];
    st[0][tid*16 + j] = p;
    st[1][tid*16 + j] = 0.0f;
  }
  __syncthreads();

  const int lane = tid & 31;
  const int wv   = tid >> 5;    // wave id 0..7
  const int mn   = lane & 15;   // row (A) / col (B,C,D) within 16x16 fragment
  const int hi   = lane >> 4;   // half-wave select per ISA fragment layout

  for (int l = 0; l < NL; ++l) {
    // --- 12 RY gates as three fused 16x16 real matrices (WMMA GEMMs) ---
    #pragma unroll
    for (int g = 0; g < 3; ++g) {
      { // Build M = R(th_{4g}) (x) R(th_{4g+1}) (x) R(th_{4g+2}) (x) R(th_{4g+3})
        int r = tid >> 4, s = tid & 15;
        float v = 1.0f;
        #pragma unroll
        for (int j = 0; j < 4; ++j) {
          int w  = l*NW + g*4 + j;
          int rb = (r >> (3 - j)) & 1, sb = (s >> (3 - j)) & 1;
          v *= (rb == sb) ? cry[w] : (sb ? -sry[w] : sry[w]);
        }
        Mmat[tid] = v;                    // Mmat[r*16 + s]
      }
      __syncthreads();

      // newS = M x S over 256 columns: 16 tiles x {re,im} = 32 wave-tasks.
      for (int it = 0; it < 4; ++it) {
        int task = wv*4 + it;             // disjoint columns per task
        int part = task & 1;
        int tile = task >> 1;
        int o    = tile*16 + mn;          // this lane's column id (B/C/D col)
        v8f acc = {0.f,0.f,0.f,0.f,0.f,0.f,0.f,0.f};
        #pragma unroll
        for (int kc = 0; kc < 4; ++kc) {  // K = 16 in chunks of 4
          int kb = kc*4 + hi*2;
          v2f afrag, bfrag;
          afrag.x = Mmat[mn*16 + kb];     // A = M[:, kb], M[:, kb+1]
          afrag.y = Mmat[mn*16 + kb + 1];
          bfrag.x = st[part][qidx(g, kb,     o)];   // B = S[kb.., col o]
          bfrag.y = st[part][qidx(g, kb + 1, o)];
          acc = __builtin_amdgcn_wmma_f32_16x16x4_f32(
              false, afrag, false, bfrag, (short)0, acc, false, false);
        }
        #pragma unroll
        for (int j = 0; j < 8; ++j)       // D rows: hi*8 + j per ISA C/D layout
          st[part][qidx(g, hi*8 + j, o)] = acc[j];
      }
      __syncthreads();
    }

    // --- 12 RZ gates fused into one diagonal phase pass ---
    for (int j = 0; j < 16; ++j) {
      int i = tid*16 + j;
      float phi = 0.0f;
      #pragma unroll
      for (int w = 0; w < NW; ++w)
        phi += ((i >> (11 - w)) & 1) ? hz[l*NW + w] : -hz[l*NW + w];
      float sp, cp;
      sincosf(phi, &sp, &cp);
      float re = st[0][i], im = st[1][i];
      st[0][i] = re*cp - im*sp;
      st[1][i] = re*sp + im*cp;
    }
    __syncthreads();

    // --- CNOT chain = GF(2)-linear index permutation, single gather pass ---
    // Forward: y_w = x_0^..^x_w (w>=1), y_0 = x_1^..^x_11.
    // Inverse (gather source): x = y ^ (y>>1) on bits 0..9, plus parity fixups.
    float tr[16], ti[16];
    for (int j = 0; j < 16; ++j) {
      int y = tid*16 + j;
      int x = (y ^ (y >> 1)) & 0x3FF;
      x |= ((((y >> 10) ^ (y >> 11) ^ y) & 1) << 10);
      x |= ((((y >> 11) ^ y) & 1) << 11);
      tr[j] = st[0][x]; ti[j] = st[1][x];
    }
    __syncthreads();
    for (int j = 0; j < 16; ++j) {
      st[0][tid*16 + j] = tr[j];
      st[1][tid*16 + j] = ti[j];
    }
    __syncthreads();
  }

  // --- <Z_w> features folded with head: out = sum_i |amp_i|^2 * c_i + b ---
  float acc = 0.0f;
  for (int j = 0; j < 16; ++j) {
    int i = tid*16 + j;
    float c = 0.0f;
    #pragma unroll
    for (int w = 0; w < NW; ++w)
      c += ((i >> (11 - w)) & 1) ? -hw[w] : hw[w];
    float re = st[0][i], im = st[1][i];
    acc += (re*re + im*im) * c;
  }
  red[tid] = acc;
  __syncthreads();
  #pragma unroll
  for (int s = 128; s > 0; s >>= 1) {
    if (tid < s) red[tid] += red[tid + s];
    __syncthreads();
  }
  if (tid == 0) out[bid] = red[0] + head_b[0];
}

extern "C" void kernel_launch(void* const* d_in, const int* in_sizes, int n_in,
                              void* d_out, int out_size, void* d_ws, size_t ws_size,
                              hipStream_t stream) {
  const float* sbatch = (const float*)d_in[0];  // state_batch (1024, 4096) f32
  const float* params = (const float*)d_in[1];  // (2, 12, 2) f32
  const float* head_w = (const float*)d_in[2];  // (1, 12) f32
  const float* head_b = (const float*)d_in[3];  // (1,) f32
  float* out = (float*)d_out;                   // (1024,) f32
  (void)in_sizes; (void)n_in; (void)out_size; (void)d_ws; (void)ws_size;
  qsim_kernel<<<1024, 256, 0, stream>>>(sbatch, params, head_w, head_b, out);
}